// CBAM_35201551958522
// MI455X (gfx1250) — compile-verified
//
#include <hip/hip_runtime.h>

#define NPTS_C     64          // output channels
#define KIN        8           // input features (cols 1..8 of points)
#define NUM_BINS   16
#define SCALE_XY   2073600     // 1440*1440
#define PID_MAX    (2*SCALE_XY)
#define BN_EPS     1e-3f
#define NEG_INF    (-3.402823466e38f)

#define SCAN_TPB   256
#define SCAN_EPT   16
#define SCAN_CHUNK (SCAN_TPB*SCAN_EPT)   // 4096 elements per scan block

typedef float v2f __attribute__((ext_vector_type(2)));
typedef float v8f __attribute__((ext_vector_type(8)));

// ---------------------------------------------------------------------------
// Workspace zeroing (rank marks, voxel counts, nvox, bn accumulators, vox sums)
// ---------------------------------------------------------------------------
__global__ void k_zero(int* __restrict__ p, long nwords) {
  long i = (long)blockIdx.x * blockDim.x + threadIdx.x;
  long stride = (long)gridDim.x * blockDim.x;
  for (; i < nwords; i += stride) p[i] = 0;
}

// ---------------------------------------------------------------------------
// Mark occupied pillar ids in the dense pillar domain
// ---------------------------------------------------------------------------
__global__ void k_mark(const int* __restrict__ pmc, int* __restrict__ rank, int n) {
  int i = blockIdx.x * blockDim.x + threadIdx.x;
  if (i >= n) return;
  int pid = pmc[i];
  pid = pid < 0 ? 0 : (pid >= PID_MAX ? PID_MAX - 1 : pid);
  rank[pid] = 1;
}

// ---------------------------------------------------------------------------
// Exclusive prefix scan over the dense pillar domain (marks -> ranks)
// ---------------------------------------------------------------------------
__global__ void k_scan_partials(const int* __restrict__ mark, int* __restrict__ partials, int total) {
  __shared__ int sh[SCAN_TPB];
  int t = threadIdx.x;
  int base = blockIdx.x * SCAN_CHUNK + t * SCAN_EPT;
  int s = 0;
  #pragma unroll
  for (int e = 0; e < SCAN_EPT; e++) {
    int i = base + e;
    if (i < total) s += mark[i];
  }
  sh[t] = s;
  __syncthreads();
  for (int off = SCAN_TPB / 2; off > 0; off >>= 1) {
    if (t < off) sh[t] += sh[t + off];
    __syncthreads();
  }
  if (t == 0) partials[blockIdx.x] = sh[0];
}

__global__ void k_scan_serial(int* __restrict__ partials, int nb) {
  if (blockIdx.x == 0 && threadIdx.x == 0) {
    int run = 0;
    for (int i = 0; i < nb; i++) { int v = partials[i]; partials[i] = run; run += v; }
  }
}

__global__ void k_scan_write(int* __restrict__ markrank, const int* __restrict__ partials, int total) {
  __shared__ int sh[SCAN_TPB];
  int t = threadIdx.x;
  int base = blockIdx.x * SCAN_CHUNK + t * SCAN_EPT;
  int vals[SCAN_EPT];
  int s = 0;
  #pragma unroll
  for (int e = 0; e < SCAN_EPT; e++) {
    int i = base + e;
    vals[e] = (i < total) ? markrank[i] : 0;
    s += vals[e];
  }
  sh[t] = s;
  __syncthreads();
  // Hillis-Steele inclusive scan of per-thread sums
  for (int off = 1; off < SCAN_TPB; off <<= 1) {
    int v = (t >= off) ? sh[t - off] : 0;
    __syncthreads();
    sh[t] += v;
    __syncthreads();
  }
  int run = partials[blockIdx.x] + (sh[t] - s);   // exclusive prefix for this thread
  #pragma unroll
  for (int e = 0; e < SCAN_EPT; e++) {
    int i = base + e;
    if (i < total) { markrank[i] = run; run += vals[e]; }
  }
}

// ---------------------------------------------------------------------------
// Per-(pillar-rank, z-bin) point counts + first-touch voxel slot allocation
// ---------------------------------------------------------------------------
__global__ void k_count(const float* __restrict__ pts, const int* __restrict__ pmc,
                        const int* __restrict__ rank, int* __restrict__ cnt_rz,
                        int* __restrict__ slotmap, int* __restrict__ nvox, int n) {
  int i = blockIdx.x * blockDim.x + threadIdx.x;
  if (i >= n) return;
  int pid = pmc[i];
  pid = pid < 0 ? 0 : (pid >= PID_MAX ? PID_MAX - 1 : pid);
  int r = rank[pid];
  float z = pts[(long)i * 9 + 6];
  int zi = (int)floorf(z);
  zi = zi < 0 ? 0 : (zi > NUM_BINS - 1 ? NUM_BINS - 1 : zi);
  int v = r * NUM_BINS + zi;
  int old = atomicAdd(&cnt_rz[v], 1);
  if (old == 0) {
    int s = atomicAdd(nvox, 1);
    slotmap[v] = s;
  }
}

// ---------------------------------------------------------------------------
// WMMA tile: 16 point rows x 64 channels, K=8 via two 16x16x4 f32 WMMA chunks
// A layout (32b 16x4): lanes0-15 row M, VGPR0=K0 / VGPR1=K1; lanes16-31 K2/K3.
// B layout (4x16): row K striped across lanes; lanes16-31 hold K+2/K+3 rows.
// C/D (16x16 f32): VGPR j = row (j + 8*laneHalf), col = lane%16.
// ---------------------------------------------------------------------------
__device__ __forceinline__ void mm_tile(const float* __restrict__ pts, int n, int base,
                                        const float* Wl, v8f acc[4]) {
  int lane = threadIdx.x & 31;
  int half = lane >> 4;
  int lm = lane & 15;
  int arow = base + lm;
  #pragma unroll
  for (int kc = 0; kc < 2; kc++) {
    int k0 = kc * 4 + 2 * half;
    v2f a;
    if (arow < n) {
      a.x = pts[(long)arow * 9 + 1 + k0];
      a.y = pts[(long)arow * 9 + 1 + k0 + 1];
    } else {
      a.x = 0.f; a.y = 0.f;
    }
    #pragma unroll
    for (int t = 0; t < 4; t++) {
      v2f b;
      b.x = Wl[(k0 + 0) * NPTS_C + t * 16 + lm];
      b.y = Wl[(k0 + 1) * NPTS_C + t * 16 + lm];
      acc[t] = __builtin_amdgcn_wmma_f32_16x16x4_f32(false, a, false, b,
                                                     (short)0, acc[t], false, false);
    }
  }
}

// ---------------------------------------------------------------------------
// Pass 1 GEMM: accumulate per-channel sum / sumsq of h = X @ W  (BN stats)
// ---------------------------------------------------------------------------
__global__ void __launch_bounds__(256) k_gemm_bnstats(const float* __restrict__ pts,
                                                      const float* __restrict__ W,
                                                      float* __restrict__ bn_sum,
                                                      float* __restrict__ bn_sq, int n) {
  __shared__ float Wl[KIN * NPTS_C];
  __shared__ float ls[NPTS_C];
  __shared__ float lq[NPTS_C];
  int tid = threadIdx.x;
  for (int i = tid; i < KIN * NPTS_C; i += 256) Wl[i] = W[i];
  if (tid < NPTS_C) { ls[tid] = 0.f; lq[tid] = 0.f; }
  __syncthreads();

  int wave = tid >> 5;
  int base = blockIdx.x * 128 + wave * 16;
  v8f acc[4] = {};
  mm_tile(pts, n, base, Wl, acc);

  int lane = tid & 31, half = lane >> 4, lm = lane & 15;
  #pragma unroll
  for (int t = 0; t < 4; t++) {
    float s = 0.f, q = 0.f;
    #pragma unroll
    for (int j = 0; j < 8; j++) {
      int row = base + half * 8 + j;
      float v = (row < n) ? acc[t][j] : 0.f;
      s += v; q += v * v;
    }
    int ch = t * 16 + lm;
    atomicAdd(&ls[ch], s);
    atomicAdd(&lq[ch], q);
  }
  __syncthreads();
  if (tid < NPTS_C) {
    atomicAdd(&bn_sum[tid], ls[tid]);
    atomicAdd(&bn_sq[tid], lq[tid]);
  }
}

// ---------------------------------------------------------------------------
// BN finalize: scale = gamma / sqrt(var+eps); shift = beta - mean*scale
// ---------------------------------------------------------------------------
__global__ void k_bn_finalize(const float* __restrict__ bn_sum, const float* __restrict__ bn_sq,
                              const float* __restrict__ gamma, const float* __restrict__ beta,
                              float* __restrict__ scale, float* __restrict__ shift, float invn) {
  int c = threadIdx.x;
  if (c >= NPTS_C) return;
  float mean = bn_sum[c] * invn;
  float var  = bn_sq[c] * invn - mean * mean;
  float inv  = 1.0f / sqrtf(var + BN_EPS);
  float sc   = gamma[c] * inv;
  scale[c] = sc;
  shift[c] = beta[c] - mean * sc;
}

// ---------------------------------------------------------------------------
// Pass 2 GEMM: affine+ReLU, add sparse_feat[pillar_rank], scatter to voxel sums
// ---------------------------------------------------------------------------
__global__ void __launch_bounds__(256) k_gemm_scatter(const float* __restrict__ pts,
                                                      const float* __restrict__ W,
                                                      const float* __restrict__ sf,
                                                      const int* __restrict__ pmc,
                                                      const int* __restrict__ rank,
                                                      const int* __restrict__ slotmap,
                                                      const float* __restrict__ scale,
                                                      const float* __restrict__ shift,
                                                      float* __restrict__ vox_sum, int n) {
  __shared__ float Wl[KIN * NPTS_C];
  __shared__ float scl[NPTS_C];
  __shared__ float shf[NPTS_C];
  __shared__ int rrank[128];
  __shared__ int rslot[128];
  int tid = threadIdx.x;
  for (int i = tid; i < KIN * NPTS_C; i += 256) Wl[i] = W[i];
  if (tid < NPTS_C) { scl[tid] = scale[tid]; shf[tid] = shift[tid]; }
  if (tid < 128) {
    int row = blockIdx.x * 128 + tid;
    int r = 0, s = 0;
    if (row < n) {
      int pid = pmc[row];
      pid = pid < 0 ? 0 : (pid >= PID_MAX ? PID_MAX - 1 : pid);
      r = rank[pid];
      float z = pts[(long)row * 9 + 6];
      int zi = (int)floorf(z);
      zi = zi < 0 ? 0 : (zi > NUM_BINS - 1 ? NUM_BINS - 1 : zi);
      s = slotmap[r * NUM_BINS + zi];
    }
    rrank[tid] = r;
    rslot[tid] = s;
  }
  __syncthreads();

  int wave = tid >> 5;
  int base = blockIdx.x * 128 + wave * 16;
  v8f acc[4] = {};
  mm_tile(pts, n, base, Wl, acc);

  int lane = tid & 31, half = lane >> 4, lm = lane & 15;
  #pragma unroll
  for (int t = 0; t < 4; t++) {
    int ch = t * 16 + lm;
    float sc = scl[ch], sh = shf[ch];
    #pragma unroll
    for (int j = 0; j < 8; j++) {
      int lrow = wave * 16 + half * 8 + j;
      int row  = blockIdx.x * 128 + lrow;
      if (row >= n) continue;
      float y = fmaxf(acc[t][j] * sc + sh, 0.f);                 // BN + ReLU
      int r = rrank[lrow], s = rslot[lrow];
      float val = sf[(long)r * NPTS_C + ch] + y;                  // + sparse_feat[inv]
      atomicAdd(&vox_sum[(long)s * NPTS_C + ch], val);
    }
  }
}

// ---------------------------------------------------------------------------
// Per-pillar max over z-bins (means), occupied gating, final output
// ---------------------------------------------------------------------------
__global__ void k_pillar_out(const float* __restrict__ sf, const int* __restrict__ cnt_rz,
                             const int* __restrict__ slotmap, const float* __restrict__ vox_sum,
                             float* __restrict__ out, int n) {
  int tid = threadIdx.x;
  int p = blockIdx.x * 4 + (tid >> 6);
  int ch = tid & 63;
  if (p >= n) return;
  float mx = NEG_INF;
  int k = 0;
  #pragma unroll
  for (int z = 0; z < NUM_BINS; z++) {
    int c = cnt_rz[p * NUM_BINS + z];
    if (c > 0) {
      k++;
      int s = slotmap[p * NUM_BINS + z];
      float mean = vox_sum[(long)s * NPTS_C + ch] / (float)c;
      mx = fmaxf(mx, mean);
    }
  }
  if (k < NUM_BINS) mx = fmaxf(mx, 0.f);   // zero-initialized unset bins join the max
  float o = sf[(long)p * NPTS_C + ch];
  if (k >= 2) o += mx;                      // occupied: pcnt >= 2
  out[(long)p * NPTS_C + ch] = o;
}

// ---------------------------------------------------------------------------
extern "C" void kernel_launch(void* const* d_in, const int* in_sizes, int n_in,
                              void* d_out, int out_size, void* d_ws, size_t ws_size,
                              hipStream_t stream) {
  const float* pts   = (const float*)d_in[0];   // [N,9]
  const float* sf    = (const float*)d_in[1];   // [N,64]
  const float* W     = (const float*)d_in[2];   // [8,64]
  const float* gamma = (const float*)d_in[3];   // [64]
  const float* beta  = (const float*)d_in[4];   // [64]
  const int*   pmc   = (const int*)d_in[5];     // [N]
  float* out = (float*)d_out;
  int n = in_sizes[5];

  // workspace layout (4-byte words)
  int* ws = (int*)d_ws;
  long off = 0;
  int*   rank     = ws + off; off += (long)PID_MAX;
  int*   cnt_rz   = ws + off; off += (long)n * NUM_BINS;
  int*   partials = ws + off; off += 2048;
  int*   nvox     = ws + off; off += 1;
  float* bn_sum   = (float*)(ws + off); off += NPTS_C;
  float* bn_sq    = (float*)(ws + off); off += NPTS_C;
  float* scale    = (float*)(ws + off); off += NPTS_C;
  float* shift    = (float*)(ws + off); off += NPTS_C;
  float* vox_sum  = (float*)(ws + off);
  long zero_words = off + (long)n * NPTS_C;     // zero everything through vox_sum
  off += (long)n * NPTS_C;
  int*   slotmap  = ws + off;                   // first-touch written; no init needed

  const int tpb = 256;
  k_zero<<<4096, tpb, 0, stream>>>(ws, zero_words);
  k_mark<<<(n + tpb - 1) / tpb, tpb, 0, stream>>>(pmc, rank, n);

  int nbscan = (PID_MAX + SCAN_CHUNK - 1) / SCAN_CHUNK;   // 1013 <= 2048
  k_scan_partials<<<nbscan, SCAN_TPB, 0, stream>>>(rank, partials, PID_MAX);
  k_scan_serial<<<1, 1, 0, stream>>>(partials, nbscan);
  k_scan_write<<<nbscan, SCAN_TPB, 0, stream>>>(rank, partials, PID_MAX);

  k_count<<<(n + tpb - 1) / tpb, tpb, 0, stream>>>(pts, pmc, rank, cnt_rz, slotmap, nvox, n);

  int nbmm = (n + 127) / 128;   // 128 point-rows per 256-thread block (8 wave32 tiles)
  k_gemm_bnstats<<<nbmm, 256, 0, stream>>>(pts, W, bn_sum, bn_sq, n);
  k_bn_finalize<<<1, 64, 0, stream>>>(bn_sum, bn_sq, gamma, beta, scale, shift, 1.0f / (float)n);
  k_gemm_scatter<<<nbmm, 256, 0, stream>>>(pts, W, sf, pmc, rank, slotmap, scale, shift, vox_sum, n);

  k_pillar_out<<<(n + 3) / 4, 256, 0, stream>>>(sf, cnt_rz, slotmap, vox_sum, out, n);
}